// upconv_layer_batch_26388279067297
// MI455X (gfx1250) — compile-verified
//
#include <hip/hip_runtime.h>

typedef __attribute__((ext_vector_type(2))) float v2f;
typedef __attribute__((ext_vector_type(8))) float v8f;

constexpr int kB      = 16;
constexpr int kCin    = 64;
constexpr int kCout   = 32;
constexpr int kO      = 224;                 // 7 * kCout
constexpr int kN      = 40962;
constexpr int kMout   = 4 * kN - 6;          // 163842
constexpr int kMTiles = kO / 16;             // 14
constexpr int kNTile  = 64;
constexpr int kNTiles = (kN + kNTile - 1) / kNTile; // 641
constexpr int kWPB    = 8;                   // waves per 256-thread block (wave32)

// ---------------------------------------------------------------------------
// Pass 1: y[b] = W @ x[b] + bias, via V_WMMA_F32_16X16X4_F32.
// One wave computes a 16(M) x 64(N) tile; K=64 -> 16 chained WMMAs per 16-col
// sub-tile. A fragments (weights) loaded once per wave and reused 4x.
// Tail columns are handled by clamped (in-bounds) loads + predicated stores
// only: B lane L affects only output column L, so padded-column garbage never
// reaches a stored element and no zero-select VALU work is needed.
// ---------------------------------------------------------------------------
__global__ __launch_bounds__(256)
void gemm_pass(const float* __restrict__ xb,   // (kCin, kN) for one batch
               const float* __restrict__ W,    // (kO, kCin) row-major
               const float* __restrict__ bias, // (kO)
               float* __restrict__ yb)         // (kO, kN)
{
  const int wave = blockIdx.x * kWPB + ((int)threadIdx.x >> 5);
  if (wave >= kMTiles * kNTiles) return;       // whole-wave uniform exit
  const int lane = (int)threadIdx.x & 31;

  const int mt   = wave % kMTiles;             // consecutive waves share x tile
  const int nt64 = wave / kMTiles;
  const int m0 = mt * 16;
  const int n0 = nt64 * kNTile;

  const int  row  = lane & 15;
  const bool hi   = lane >= 16;
  const int  koff = hi ? 2 : 0;                // A: V0->K0/K2, V1->K1/K3 split

  // Preload 16 A-fragments (16x4 each) covering K = 0..63.
  v2f a[16];
#pragma unroll
  for (int kk = 0; kk < 16; ++kk) {
    const float* wp = W + (m0 + row) * kCin + kk * 4 + koff;
    a[kk].x = wp[0];
    a[kk].y = wp[1];
  }

  // Bias by output row (same value across the 16 lanes of each half-wave).
  float binit[8];
#pragma unroll
  for (int v = 0; v < 8; ++v)
    binit[v] = bias[m0 + v + (hi ? 8 : 0)];

#pragma unroll
  for (int nt = 0; nt < 4; ++nt) {
    const int  ncol  = n0 + nt * 16 + row;
    const bool valid = ncol < kN;
    const int  nc    = valid ? ncol : (kN - 1); // clamp: loads stay in-bounds
                                                // and EXEC stays all-ones

    // C/D layout: acc[v] <-> output row (m0 + v) for lanes 0-15, (m0+v+8) hi.
    v8f acc;
#pragma unroll
    for (int v = 0; v < 8; ++v)
      acc[v] = binit[v];

#pragma unroll
    for (int kk = 0; kk < 16; ++kk) {
      const int c0 = kk * 4 + koff;            // B: same K split as A
      v2f bf;
      bf.x = xb[(long)c0 * kN + nc];
      bf.y = xb[(long)(c0 + 1) * kN + nc];
      // (neg_a, A, neg_b, B, c_mod, C, reuse_a, reuse_b)
      acc = __builtin_amdgcn_wmma_f32_16x16x4_f32(false, a[kk], false, bf,
                                                  (short)0, acc, false, false);
    }

    if (valid) {
#pragma unroll
      for (int v = 0; v < 8; ++v) {
        const int o = m0 + v + (hi ? 8 : 0);
        yb[(long)o * kN + ncol] = acc[v];
      }
    }
  }
}

// ---------------------------------------------------------------------------
// Pass 2: fused gather + pair-average from the (co, 7N) view of y[b].
// m-fastest thread mapping -> coalesced NT stores; gathered reads hit the
// L2-resident y[b] slice (concurrent blocks share one co -> ~1.1 MB hot set).
// ---------------------------------------------------------------------------
__global__ __launch_bounds__(256)
void gather_pass(const float* __restrict__ yb,   // (kO, kN)
                 const int*   __restrict__ top,  // (kN)
                 const int*   __restrict__ down, // (2*(3N-6))
                 float*       __restrict__ outb) // (kCout, kMout)
{
  const long e = (long)blockIdx.x * blockDim.x + threadIdx.x;
  if (e >= (long)kCout * kMout) return;
  const int co = (int)(e / kMout);
  const int m  = (int)(e % kMout);
  const float* ybase = yb + (long)co * 7 * kN;

  float val;
  if (m < kN) {
    const unsigned j = (unsigned)top[m];
    const unsigned k = j / (unsigned)kN;          // const-divisor -> mulhi
    const unsigned n = j - k * (unsigned)kN;
    val = ybase[(long)k * kN + n];
  } else {
    const int t = m - kN;
    const unsigned j0 = (unsigned)down[2 * t];
    const unsigned j1 = (unsigned)down[2 * t + 1];
    const unsigned k0 = j0 / (unsigned)kN, n0 = j0 - k0 * (unsigned)kN;
    const unsigned k1 = j1 / (unsigned)kN, n1 = j1 - k1 * (unsigned)kN;
    val = 0.5f * (ybase[(long)k0 * kN + n0] + ybase[(long)k1 * kN + n1]);
  }
  __builtin_nontemporal_store(val, &outb[e]);     // output never re-read
}

// ---------------------------------------------------------------------------
extern "C" void kernel_launch(void* const* d_in, const int* in_sizes, int n_in,
                              void* d_out, int out_size, void* d_ws, size_t ws_size,
                              hipStream_t stream) {
  (void)in_sizes; (void)n_in; (void)out_size; (void)ws_size;
  const float* x    = (const float*)d_in[0];
  const float* W    = (const float*)d_in[1];
  const float* bias = (const float*)d_in[2];
  const int*   top  = (const int*)d_in[3];
  const int*   down = (const int*)d_in[4];
  float* out = (float*)d_out;
  float* yws = (float*)d_ws;   // needs kO*kN*4 ~ 35 MB (L2-resident)

  const int gemmWaves  = kMTiles * kNTiles;
  const int gemmBlocks = (gemmWaves + kWPB - 1) / kWPB;
  const long gatherElems = (long)kCout * kMout;
  const int gatherBlocks = (int)((gatherElems + 255) / 256);

  // Per-batch pipeline: in-order stream serializes gemm(b) -> gather(b), so a
  // single 35 MB y buffer is reused and stays hot in the 192 MB L2.
  for (int b = 0; b < kB; ++b) {
    const float* xb   = x   + (long)b * kCin  * kN;
    float*       outb = out + (long)b * kCout * kMout;
    gemm_pass  <<<gemmBlocks,   256, 0, stream>>>(xb, W, bias, yws);
    gather_pass<<<gatherBlocks, 256, 0, stream>>>(yws, top, down, outb);
  }
}